// CrossAttention_18287970746581
// MI455X (gfx1250) — compile-verified
//
#include <hip/hip_runtime.h>
#include <hip/hip_bf16.h>

// CDNA5 / gfx1250: wave32, WMMA f32 16x16x4, async global->LDS staging.
typedef __attribute__((ext_vector_type(2))) float v2f;
typedef __attribute__((ext_vector_type(8))) float v8f;

#define N_HEADS 6
#define D_DIM   32
#define FEAT    16
#define HID     (N_HEADS * D_DIM)   // 192
#define BATCH   64
#define MAX_D   50
#define MAX_P   545
#define LD_PAD  64                   // 50 -> 64
#define LP_PAD  560                  // 545 -> 560

// ---------------------------------------------------------------------------
// Generic row-major f32 WMMA GEMM: C[M,N] = A[M,K] @ B[K,N] (+ resid[M,N]).
// One wave (32 threads) computes one 16x16 C tile via K/4 v_wmma_f32_16x16x4_f32.
// Requires M%16==0, N%16==0, K%4==0 (true for every call site here).
// ---------------------------------------------------------------------------
__global__ void wmma_gemm_f32(const float* __restrict__ A, int lda,
                              const float* __restrict__ Bm, int ldb,
                              float* __restrict__ C, int ldc,
                              const float* __restrict__ resid, int ldr,
                              int K) {
  const int tileN = blockIdx.x * 16;
  const int tileM = blockIdx.y * 16;
  const int lane  = threadIdx.x & 31;
  const int half  = lane >> 4;     // selects K pair (A/B) and row half (C)
  const int lidx  = lane & 15;

  v8f acc = {};
  const float* ap = A + (long)(tileM + lidx) * lda;
  for (int k0 = 0; k0 < K; k0 += 4) {
    const int kk = k0 + half * 2;
    v2f a = *(const v2f*)(ap + kk);                       // A[m][kk..kk+1]
    v2f b;
    const float* bp = Bm + (long)kk * ldb + tileN + lidx; // B[kk][n], B[kk+1][n]
    b.x = bp[0];
    b.y = bp[ldb];
    acc = __builtin_amdgcn_wmma_f32_16x16x4_f32(false, a, false, b,
                                                (short)0, acc, false, false);
  }
  const int row0 = tileM + half * 8;
  const int col  = tileN + lidx;
  #pragma unroll
  for (int i = 0; i < 8; ++i) {
    float v = acc[i];
    const int r = row0 + i;
    if (resid) v += resid[(long)r * ldr + col];
    C[(long)r * ldc + col] = v;
  }
}

// ---------------------------------------------------------------------------
// Fused attention for one (batch, head) per workgroup.
// q/k/v/ctx are [B*L, 192] row-major, head h occupies columns h*32..h*32+31.
// LDS: K_bh [LkPad,32] + V_bh [LkPad,32] shared by all waves,
//      plus per-wave score rows [16, LkPad].  (Protein: 287 KB < 320 KB WGP LDS.)
// K/V staged with GLOBAL_LOAD_ASYNC_TO_LDS_B128 (ASYNCcnt path, no VGPR bounce).
// ---------------------------------------------------------------------------
__global__ void attention_f32(const float* __restrict__ q,
                              const float* __restrict__ kmat,
                              const float* __restrict__ vmat,
                              float* __restrict__ ctx,
                              int Lq, int Lk, int LkPad) {
  extern __shared__ float smem[];
  const int b    = blockIdx.x / N_HEADS;
  const int h    = blockIdx.x % N_HEADS;
  const int tid  = threadIdx.x;
  const int lane = tid & 31;
  const int wave = tid >> 5;
  const int nwaves = blockDim.x >> 5;

  float* Ks  = smem;
  float* Vs  = smem + (long)LkPad * 32;
  float* Ssc = smem + (long)LkPad * 64 + (long)wave * 16 * LkPad;

  // Zero the padded tail rows with plain ds stores (disjoint from async dest).
  for (int idx = tid; idx < (LkPad - Lk) * 32; idx += blockDim.x) {
    Ks[Lk * 32 + idx] = 0.f;
    Vs[Lk * 32 + idx] = 0.f;
  }

  // Async global->LDS copy of the real K/V rows, 16 bytes per lane-op.
  const long kvBase = ((long)b * Lk) * HID + h * D_DIM;
  const int nvec = Lk * (D_DIM / 4);            // 16-byte chunks per matrix
  for (int i = tid; i < nvec; i += blockDim.x) {
    const int l  = i >> 3;
    const int d4 = (i & 7) * 4;
    const float* gk = kmat + kvBase + (long)l * HID + d4;
    const float* gv = vmat + kvBase + (long)l * HID + d4;
    const unsigned ldsK = (unsigned)(size_t)(const void*)(Ks + l * 32 + d4);
    const unsigned ldsV = (unsigned)(size_t)(const void*)(Vs + l * 32 + d4);
    asm volatile("global_load_async_to_lds_b128 %0, %1, off"
                 :: "v"(ldsK), "v"(gk) : "memory");
    asm volatile("global_load_async_to_lds_b128 %0, %1, off"
                 :: "v"(ldsV), "v"(gv) : "memory");
  }
  asm volatile("s_wait_asynccnt 0x0" ::: "memory");
  __syncthreads();

  const int half = lane >> 4;
  const int lidx = lane & 15;
  const int nQTiles = (Lq + 15) / 16;
  const float scale = 0.17677669529663687f;   // 1/sqrt(32)
  const long qBase = ((long)b * Lq) * HID + h * D_DIM;

  for (int qt = wave; qt < nQTiles; qt += nwaves) {
    // Preload this q tile's A-fragments (16x32 -> 8 WMMA K-steps).
    v2f qfrag[8];
    const int m = qt * 16 + lidx;
    const bool mok = (m < Lq);
    #pragma unroll
    for (int s = 0; s < 8; ++s) {
      const int kk = s * 4 + half * 2;
      v2f qv = {};
      if (mok) qv = *(const v2f*)(q + qBase + (long)m * HID + kk);
      qfrag[s] = qv;
    }

    // scores[16, LkPad] = (q_tile @ K^T) * scale  -> per-wave LDS rows
    for (int kt = 0; kt < LkPad / 16; ++kt) {
      v8f acc = {};
      #pragma unroll
      for (int s = 0; s < 8; ++s) {
        const int kk = s * 4 + half * 2;
        // B[k][n] = K[kt*16+n][k] (transposed 8B read from LDS)
        v2f bfrag = *(const v2f*)(Ks + (kt * 16 + lidx) * 32 + kk);
        acc = __builtin_amdgcn_wmma_f32_16x16x4_f32(false, qfrag[s], false, bfrag,
                                                    (short)0, acc, false, false);
      }
      #pragma unroll
      for (int i = 0; i < 8; ++i) {
        const int r = half * 8 + i;
        Ssc[r * LkPad + kt * 16 + lidx] = acc[i] * scale;
      }
    }

    // Row softmax over the true Lk columns, all 32 lanes active:
    // lane pair (r, r+16) splits row r; halves combined via wave32 shfl_xor(16).
    {
      float* row = Ssc + lidx * LkPad;
      const int cmid = (Lk + 1) >> 1;
      const int c0 = half ? cmid : 0;
      const int c1 = half ? Lk : cmid;
      float mx = -1e30f;
      for (int c = c0; c < c1; ++c) mx = fmaxf(mx, row[c]);
      mx = fmaxf(mx, __shfl_xor(mx, 16, 32));
      float sum = 0.f;
      for (int c = c0; c < c1; ++c) { const float e = expf(row[c] - mx); row[c] = e; sum += e; }
      sum += __shfl_xor(sum, 16, 32);
      const float inv = 1.0f / sum;
      for (int c = c0; c < c1; ++c) row[c] *= inv;
    }
    // Zero padded columns (16 rows x (LkPad-Lk) cols), all lanes.
    {
      const int padw = LkPad - Lk;
      for (int idx = lane; idx < 16 * padw; idx += 32) {
        const int r = idx / padw, c = Lk + idx % padw;
        Ssc[r * LkPad + c] = 0.f;
      }
    }

    // ctx[16,32] = P[16,LkPad] @ V[LkPad,32]
    #pragma unroll
    for (int ct = 0; ct < 2; ++ct) {
      v8f acc = {};
      for (int kk = 0; kk < LkPad; kk += 4) {
        const int kd = kk + half * 2;
        v2f a = *(const v2f*)(Ssc + lidx * LkPad + kd);   // P in A-matrix layout
        v2f bv;
        bv.x = Vs[kd * 32 + ct * 16 + lidx];
        bv.y = Vs[(kd + 1) * 32 + ct * 16 + lidx];
        acc = __builtin_amdgcn_wmma_f32_16x16x4_f32(false, a, false, bv,
                                                    (short)0, acc, false, false);
      }
      #pragma unroll
      for (int i = 0; i < 8; ++i) {
        const int r = qt * 16 + half * 8 + i;
        if (r < Lq)
          ctx[((long)b * Lq + r) * HID + h * D_DIM + ct * 16 + lidx] = acc[i];
      }
    }
  }
}

// ---------------------------------------------------------------------------
// pooled = max over MAX_D rows of out3[B*MAX_D,16]; 3-layer MLP; sigmoid.
// ---------------------------------------------------------------------------
__global__ void pool_mlp_kernel(const float* __restrict__ out3,
                                const float* __restrict__ d1w, const float* __restrict__ d1b,
                                const float* __restrict__ d2w, const float* __restrict__ d2b,
                                const float* __restrict__ d3w, const float* __restrict__ d3b,
                                float* __restrict__ out) {
  const int bidx = blockIdx.x * blockDim.x + threadIdx.x;
  if (bidx >= BATCH) return;
  float pooled[FEAT];
  #pragma unroll
  for (int f = 0; f < FEAT; ++f) pooled[f] = -1e30f;
  for (int l = 0; l < MAX_D; ++l) {
    const float* row = out3 + ((long)bidx * MAX_D + l) * FEAT;
    #pragma unroll
    for (int f = 0; f < FEAT; ++f) pooled[f] = fmaxf(pooled[f], row[f]);
  }
  float h1[32];
  for (int j = 0; j < 32; ++j) {
    float s = d1b[j];
    for (int f = 0; f < FEAT; ++f) s += pooled[f] * d1w[f * 32 + j];
    h1[j] = fmaxf(s, 0.f);
  }
  float h2[16];
  for (int j = 0; j < 16; ++j) {
    float s = d2b[j];
    for (int f = 0; f < 32; ++f) s += h1[f] * d2w[f * 16 + j];
    h2[j] = fmaxf(s, 0.f);
  }
  float s = d3b[0];
  for (int f = 0; f < 16; ++f) s += h2[f] * d3w[f];
  out[bidx] = 1.0f / (1.0f + expf(-s));
}

// ---------------------------------------------------------------------------
extern "C" void kernel_launch(void* const* d_in, const int* in_sizes, int n_in,
                              void* d_out, int out_size, void* d_ws, size_t ws_size,
                              hipStream_t stream) {
  (void)in_sizes; (void)n_in; (void)out_size; (void)ws_size;

  // Inputs in setup_inputs() dict order. Embedding/mask inputs (0-3, 6-13)
  // are dead code w.r.t. the output and are intentionally unused.
  const float* in3   = (const float*)d_in[4];    // [64,50,16]
  const float* in4   = (const float*)d_in[5];    // [64,545,16]
  const float* a1_wq = (const float*)d_in[14];
  const float* a1_wk = (const float*)d_in[15];
  const float* a1_wv = (const float*)d_in[16];
  const float* a1_wo = (const float*)d_in[17];
  const float* a2_wq = (const float*)d_in[18];
  const float* a2_wk = (const float*)d_in[19];
  const float* a2_wv = (const float*)d_in[20];
  const float* a2_wo = (const float*)d_in[21];
  const float* d1w = (const float*)d_in[22];
  const float* d1b = (const float*)d_in[23];
  const float* d2w = (const float*)d_in[24];
  const float* d2b = (const float*)d_in[25];
  const float* d3w = (const float*)d_in[26];
  const float* d3b = (const float*)d_in[27];
  float* out = (float*)d_out;

  const int MD = BATCH * MAX_D;   // 3200  (multiple of 16)
  const int MP = BATCH * MAX_P;   // 34880 (multiple of 16)

  // Workspace carve-up (~120 MB of f32; whole working set fits in 192 MB L2).
  float* ws = (float*)d_ws;
  size_t off = 0;
  auto alloc = [&](size_t n) { float* p = ws + off; off += n; return p; };
  float* q1   = alloc((size_t)MD * HID);
  float* k1   = alloc((size_t)MD * HID);
  float* v1   = alloc((size_t)MD * HID);
  float* q2   = alloc((size_t)MP * HID);
  float* k2   = alloc((size_t)MP * HID);
  float* v2   = alloc((size_t)MP * HID);
  float* ctx1 = alloc((size_t)MD * HID);
  float* ctx2 = alloc((size_t)MP * HID);
  float* out1 = alloc((size_t)MD * FEAT);
  float* out2 = alloc((size_t)MP * FEAT);
  float* out3 = alloc((size_t)MD * FEAT);

  const dim3 blk32(32);
  const dim3 gProjD(HID / 16, MD / 16);   // [3200,16]@[16,192]
  const dim3 gProjP(HID / 16, MP / 16);   // [34880,16]@[16,192]
  const dim3 gOutD(FEAT / 16, MD / 16);   // [3200,192]@[192,16]
  const dim3 gOutP(FEAT / 16, MP / 16);

  // --- QKV projections (WMMA GEMMs) ---
  wmma_gemm_f32<<<gProjD, blk32, 0, stream>>>(in3, FEAT, a1_wq, HID, q1, HID, nullptr, 0, FEAT);
  wmma_gemm_f32<<<gProjD, blk32, 0, stream>>>(in3, FEAT, a1_wk, HID, k1, HID, nullptr, 0, FEAT);
  wmma_gemm_f32<<<gProjD, blk32, 0, stream>>>(in3, FEAT, a1_wv, HID, v1, HID, nullptr, 0, FEAT);
  wmma_gemm_f32<<<gProjP, blk32, 0, stream>>>(in4, FEAT, a2_wq, HID, q2, HID, nullptr, 0, FEAT);
  wmma_gemm_f32<<<gProjP, blk32, 0, stream>>>(in4, FEAT, a2_wk, HID, k2, HID, nullptr, 0, FEAT);
  wmma_gemm_f32<<<gProjP, blk32, 0, stream>>>(in4, FEAT, a2_wv, HID, v2, HID, nullptr, 0, FEAT);

  // --- Self attentions (fused WMMA + async-staged LDS-resident K/V) ---
  const dim3 gAttn(BATCH * N_HEADS);
  const dim3 blk128(128);
  const size_t smemD = (size_t)(LD_PAD * 64 + 4 * 16 * LD_PAD) * sizeof(float);  //  32 KB
  const size_t smemP = (size_t)(LP_PAD * 64 + 4 * 16 * LP_PAD) * sizeof(float);  // 280 KB (<320 KB WGP LDS)
  attention_f32<<<gAttn, blk128, smemD, stream>>>(q1, k1, v1, ctx1, MAX_D, MAX_D, LD_PAD);
  attention_f32<<<gAttn, blk128, smemP, stream>>>(q2, k2, v2, ctx2, MAX_P, MAX_P, LP_PAD);

  // --- Output projections + residual ---
  wmma_gemm_f32<<<gOutD, blk32, 0, stream>>>(ctx1, HID, a1_wo, FEAT, out1, FEAT, in3, FEAT, HID);
  wmma_gemm_f32<<<gOutP, blk32, 0, stream>>>(ctx2, HID, a2_wo, FEAT, out2, FEAT, in4, FEAT, HID);

  // --- Cross attention: q from out1, k/v from out2 (a2 weights); reuse buffers ---
  wmma_gemm_f32<<<gProjD, blk32, 0, stream>>>(out1, FEAT, a2_wq, HID, q1, HID, nullptr, 0, FEAT);
  wmma_gemm_f32<<<gProjP, blk32, 0, stream>>>(out2, FEAT, a2_wk, HID, k2, HID, nullptr, 0, FEAT);
  wmma_gemm_f32<<<gProjP, blk32, 0, stream>>>(out2, FEAT, a2_wv, HID, v2, HID, nullptr, 0, FEAT);
  attention_f32<<<gAttn, blk128, smemP, stream>>>(q1, k2, v2, ctx1, MAX_D, MAX_P, LP_PAD);
  wmma_gemm_f32<<<gOutD, blk32, 0, stream>>>(ctx1, HID, a2_wo, FEAT, out3, FEAT, out1, FEAT, HID);

  // --- Max-pool + MLP + sigmoid ---
  pool_mlp_kernel<<<1, 64, 0, stream>>>(out3, d1w, d1b, d2w, d2b, d3w, d3b, out);
}